// GraphSAGEEncoder_34600256537001
// MI455X (gfx1250) — compile-verified
//
#include <hip/hip_runtime.h>
#include <math.h>

typedef float v2f __attribute__((ext_vector_type(2)));
typedef float v8f __attribute__((ext_vector_type(8)));

#define DIN 128
#define EPS_BN 1e-5f
#define EPS_NORM 1e-12f

// ---------------------------------------------------------------- utilities
__global__ void zero_f32(float* __restrict__ p, long n) {
    long i = (long)blockIdx.x * blockDim.x + threadIdx.x;
    long stride = (long)gridDim.x * blockDim.x;
    for (; i < n; i += stride) p[i] = 0.0f;
}

__global__ void degree_count(const int* __restrict__ dst, float* __restrict__ deg, int E) {
    int e = blockIdx.x * blockDim.x + threadIdx.x;
    if (e < E) atomicAdd(&deg[dst[e]], 1.0f);
}

__global__ void inv_degree(const float* __restrict__ deg, float* __restrict__ invdeg, int n) {
    int i = blockIdx.x * blockDim.x + threadIdx.x;
    if (i < n) invdeg[i] = 1.0f / fmaxf(deg[i], 1.0f);
}

// -------------------------------------------------------- edge scatter (din=128)
// one thread = one (edge, 4-feature chunk); 32 consecutive threads cover one edge
__global__ void scatter_add128(const float* __restrict__ h, const int* __restrict__ src,
                               const int* __restrict__ dst, float* __restrict__ agg, int E) {
    int t = blockIdx.x * blockDim.x + threadIdx.x;
    if (t >= (E << 5)) return;
    int e = t >> 5;
    int c = (t & 31) << 2;            // feature offset 0..124
    int s = src[e];
    int d = dst[e];
    const float4 v = *(const float4*)(h + (long)s * DIN + c);
    float* p = agg + (long)d * DIN + c;
    atomicAdd(p + 0, v.x);
    atomicAdd(p + 1, v.y);
    atomicAdd(p + 2, v.z);
    atomicAdd(p + 3, v.w);
}

// -------------------------------------------- fused SAGEConv GEMM + BN (+ReLU)
// out[m,n] = BN( mean[m,:]·Wl[n,:] + bl[n] + h[m,:]·Wr[n,:] )
// one wave -> 16x16 output tile; block = (dout/16) waves covering 16 rows x dout cols.
// The block's 16x128 agg/h tiles are contiguous 8KB slabs -> staged into LDS once
// via gfx1250 async-to-LDS DMA and shared by all waves.
__global__ void sage_gemm(const float* __restrict__ agg, const float* __restrict__ invdeg,
                          const float* __restrict__ h,
                          const float* __restrict__ Wl, const float* __restrict__ bl,
                          const float* __restrict__ Wr,
                          const float* __restrict__ gam, const float* __restrict__ bet,
                          const float* __restrict__ rm, const float* __restrict__ rv,
                          float* __restrict__ out, int dout, int relu) {
    __shared__ float s_agg[16 * DIN];   // 8 KB
    __shared__ float s_h[16 * DIN];     // 8 KB

    const int tid = threadIdx.x;
    const int row_base = blockIdx.x * 16;

    // ---- stage A tiles: contiguous [16 x 128] slabs of agg and h -> LDS -------
    {
        const float* gA = agg + (long)row_base * DIN;
        const float* gH = h   + (long)row_base * DIN;
        for (int j = tid; j < (16 * DIN) / 4; j += blockDim.x) {
            unsigned ldsA = (unsigned)(unsigned long long)(const void*)&s_agg[j * 4];
            unsigned ldsH = (unsigned)(unsigned long long)(const void*)&s_h[j * 4];
            int goff = j * 16;  // byte offset into the slab
            asm volatile("global_load_async_to_lds_b128 %0, %1, %2"
                         :: "v"(ldsA), "v"(goff), "s"(gA) : "memory");
            asm volatile("global_load_async_to_lds_b128 %0, %1, %2"
                         :: "v"(ldsH), "v"(goff), "s"(gH) : "memory");
        }
        asm volatile("s_wait_asynccnt 0x0" ::: "memory");
        __syncthreads();
    }

    const int lane = tid & 31;
    const int wave = tid >> 5;                   // n-tile within block
    const int n_base = wave * 16;

    const int half = lane >> 4;                  // 0: K pair base 0, 1: base 2
    const int lmod = lane & 15;
    const int bcol = n_base + lmod;              // B-matrix column held by this lane

    const float idg = invdeg[row_base + lmod];

    const float* lagg = s_agg + lmod * DIN;      // A rows now in LDS
    const float* lh   = s_h   + lmod * DIN;
    const float* wlp  = Wl + (long)bcol * DIN;   // B[k][n] = Wl[n][k]
    const float* wrp  = Wr + (long)bcol * DIN;

    v8f acc = {};
#pragma unroll 4
    for (int k = 0; k < DIN; k += 4) {
        const int k0 = k + half * 2;             // ISA 16x4 f32 A layout
        v2f a_mean = *(const v2f*)(lagg + k0);
        a_mean *= idg;
        v2f a_root = *(const v2f*)(lh + k0);
        v2f b_l = *(const v2f*)(wlp + k0);
        v2f b_r = *(const v2f*)(wrp + k0);
        acc = __builtin_amdgcn_wmma_f32_16x16x4_f32(false, a_mean, false, b_l,
                                                    (short)0, acc, false, false);
        acc = __builtin_amdgcn_wmma_f32_16x16x4_f32(false, a_root, false, b_r,
                                                    (short)0, acc, false, false);
    }

    // fold bias + eval-mode BN into scale/shift per output column
    const int col = bcol;
    const float sc = gam[col] * rsqrtf(rv[col] + EPS_BN);
    const float sh = (bl[col] - rm[col]) * sc + bet[col];

#pragma unroll
    for (int v = 0; v < 8; ++v) {
        const int m = v + 8 * half;              // ISA 16x16 f32 C/D layout
        float val = acc[v] * sc + sh;
        if (relu) val = fmaxf(val, 0.0f);
        out[(long)(row_base + m) * dout + col] = val;
    }
}

// ------------------------------------------------- final rowwise L2 normalize (dout=64)
__global__ void l2norm64(float* __restrict__ out, int n) {
    int node = blockIdx.x * (blockDim.x >> 5) + (threadIdx.x >> 5);
    int lane = threadIdx.x & 31;
    if (node >= n) return;
    float a = out[(long)node * 64 + lane];
    float b = out[(long)node * 64 + lane + 32];
    float s = a * a + b * b;
#pragma unroll
    for (int off = 16; off > 0; off >>= 1) s += __shfl_xor(s, off, 32);
    float inv = 1.0f / fmaxf(sqrtf(s), EPS_NORM);
    out[(long)node * 64 + lane]      = a * inv;
    out[(long)node * 64 + lane + 32] = b * inv;
}

// ---------------------------------------------------------------- launcher
extern "C" void kernel_launch(void* const* d_in, const int* in_sizes, int n_in,
                              void* d_out, int out_size, void* d_ws, size_t ws_size,
                              hipStream_t stream) {
    const float* x  = (const float*)d_in[0];
    const int*   ei = (const int*)d_in[1];
    const int Nn = in_sizes[0] / DIN;        // 50000
    const int Ee = in_sizes[1] / 2;          // 800000
    const int* src = ei;                     // edge_index[0]
    const int* dst = ei + Ee;                // edge_index[1]

    // per-layer params: Wl, bl, Wr, gamma, beta, running_mean, running_var
    const float* P[3][7];
    for (int i = 0; i < 3; ++i)
        for (int j = 0; j < 7; ++j)
            P[i][j] = (const float*)d_in[2 + i * 7 + j];

    float* ws     = (float*)d_ws;
    float* agg    = ws;                                  // [N,128]
    float* hA     = agg + (size_t)Nn * DIN;              // [N,128]
    float* hB     = hA  + (size_t)Nn * DIN;              // [N,128]
    float* deg    = hB  + (size_t)Nn * DIN;              // [N]
    float* invdeg = deg + Nn;                            // [N]

    // degree (graph is fixed across layers)
    zero_f32<<<256, 256, 0, stream>>>(deg, Nn);
    degree_count<<<(Ee + 255) / 256, 256, 0, stream>>>(dst, deg, Ee);
    inv_degree<<<(Nn + 255) / 256, 256, 0, stream>>>(deg, invdeg, Nn);

    const float* hin = x;
    float* houts[3]     = { hA, hB, (float*)d_out };
    const int  douts[3] = { 128, 128, 64 };

    for (int L = 0; L < 3; ++L) {
        zero_f32<<<2048, 256, 0, stream>>>(agg, (long)Nn * DIN);
        scatter_add128<<<((Ee << 5) + 255) / 256, 256, 0, stream>>>(hin, src, dst, agg, Ee);

        const int dout = douts[L];
        dim3 blk(32 * (dout / 16));          // 8 waves (256) or 4 waves (128)
        dim3 grd(Nn / 16);                   // 3125 row tiles, exact
        sage_gemm<<<grd, blk, 0, stream>>>(agg, invdeg, hin,
                                           P[L][0], P[L][1], P[L][2],
                                           P[L][3], P[L][4], P[L][5], P[L][6],
                                           houts[L], dout, (L < 2) ? 1 : 0);
        hin = houts[L];
    }

    l2norm64<<<(Nn + 7) / 8, 256, 0, stream>>>((float*)d_out, Nn);
}